// QueryConditionedTransportScorerV4_15934328668555
// MI455X (gfx1250) — compile-verified
//
#include <hip/hip_runtime.h>
#include <math.h>

#define EPSF     1e-8f
#define LN_EPSF  1e-5f
#define SINK_EPS 0.05f
#define SINK_ITERS 60
// exp((x)/eps) == exp2(x * K2E),  eps*ln(s) == CLG * log2(s)
#define K2E  28.853900817779268f   // (1/0.05) * log2(e)
#define CLG  0.034657359027997264f // 0.05 * ln(2)

typedef float v2f __attribute__((ext_vector_type(2)));
typedef float v8f __attribute__((ext_vector_type(8)));

// ---------------------------------------------------------------------------
// CDNA5 fp32 WMMA: D(16x16) = A(16x4) * B(4x16) + C   (wave32, even VGPRs)
// A frag: lane l -> M = l%16, VGPR0 = A[M][2*(l/16)],   VGPR1 = A[M][2*(l/16)+1]
// B frag: lane l -> N = l%16, VGPR0 = B[2*(l/16)][N],   VGPR1 = B[2*(l/16)+1][N]
// C/D   : VGPR r -> M = r + 8*(l/16), N = l%16
// ---------------------------------------------------------------------------
__device__ __forceinline__ v8f wmma_f32_16x16x4(v2f a, v2f b, v8f c) {
  return __builtin_amdgcn_wmma_f32_16x16x4_f32(false, a, false, b, (short)0, c,
                                               false, false);
}

__device__ __forceinline__ float block_reduce_sum(float v, float* red, int t, int n) {
  red[t] = v; __syncthreads();
  for (int s = n >> 1; s > 0; s >>= 1) {
    if (t < s) red[t] += red[t + s];
    __syncthreads();
  }
  float r = red[0]; __syncthreads();
  return r;
}

__device__ __forceinline__ float gelu_exact(float x) {
  return 0.5f * x * (1.0f + erff(x * 0.70710678118654752f));
}

// ---------------------------------------------------------------------------
// Stage 1: per-query mass-normalized pooling + LayerNorm  -> pooled_ln[256,128]
// ---------------------------------------------------------------------------
__global__ void k_pool(const float* __restrict__ qatoms,   // [256,128,128]
                       const float* __restrict__ qmass,    // [256,128]
                       const float* __restrict__ g,        // [128]
                       const float* __restrict__ b,        // [128]
                       float* __restrict__ out) {          // [256,128]
  const int q = blockIdx.x, t = threadIdx.x;   // 128 threads
  __shared__ float sm[128];
  __shared__ float red[128];

  float mraw = fmaxf(qmass[q * 128 + t], 0.0f);
  float msum = block_reduce_sum(mraw, red, t, 128);
  sm[t] = mraw / fmaxf(msum, EPSF);
  __syncthreads();

  const float* base = qatoms + q * 128 * 128;
  float a0 = 0.0f, a1 = 0.0f;
  for (int m = 0; m < 128; m += 2) {
    a0 = fmaf(sm[m],     base[m * 128 + t],       a0);
    a1 = fmaf(sm[m + 1], base[(m + 1) * 128 + t], a1);
  }
  float acc = a0 + a1;
  float mean = block_reduce_sum(acc, red, t, 128) * (1.0f / 128.0f);
  float d = acc - mean;
  float var = block_reduce_sum(d * d, red, t, 128) * (1.0f / 128.0f);
  out[q * 128 + t] = d * rsqrtf(var + LN_EPSF) * g[t] + b[t];
}

// ---------------------------------------------------------------------------
// Stage 2: context concat + LayerNorm (padded to 16 rows) -> ctx_ln[16,512]
// ---------------------------------------------------------------------------
__global__ void k_ctx(const float* __restrict__ csum,  // [10,256]
                      const float* __restrict__ ectx,  // [10,256]
                      const float* __restrict__ g,     // [512]
                      const float* __restrict__ b,     // [512]
                      float* __restrict__ out) {       // [16,512]
  const int w = blockIdx.x, t = threadIdx.x;   // 512 threads
  __shared__ float red[512];
  if (w >= 10) { out[w * 512 + t] = 0.0f; return; }
  float v = (t < 256) ? csum[w * 256 + t] : ectx[w * 256 + (t - 256)];
  float mean = block_reduce_sum(v, red, t, 512) * (1.0f / 512.0f);
  float d = v - mean;
  float var = block_reduce_sum(d * d, red, t, 512) * (1.0f / 512.0f);
  out[w * 512 + t] = d * rsqrtf(var + LN_EPSF) * g[t] + b[t];
}

// ---------------------------------------------------------------------------
// Stage 3: fused pair features + LayerNorm -> fused_ln[2560,1024]
// ---------------------------------------------------------------------------
__global__ void k_fused(const float* __restrict__ qh,   // [256,256]
                        const float* __restrict__ ch,   // [16,256]
                        const float* __restrict__ g,    // [1024]
                        const float* __restrict__ b,    // [1024]
                        float* __restrict__ out) {      // [2560,1024]
  const int p = blockIdx.x, t = threadIdx.x;   // 256 threads
  const int q = p / 10, w = p % 10;
  __shared__ float buf[1024];
  __shared__ float red[256];
  float qv = qh[q * 256 + t];
  float cv = ch[w * 256 + t];
  buf[t]        = qv;
  buf[256 + t]  = cv;
  buf[512 + t]  = qv * cv;
  buf[768 + t]  = fabsf(qv - cv);
  __syncthreads();
  float s = (buf[t] + buf[256 + t]) + (buf[512 + t] + buf[768 + t]);
  float mean = block_reduce_sum(s, red, t, 256) * (1.0f / 1024.0f);
  float sq = 0.0f;
  for (int j = 0; j < 4; ++j) { float d = buf[j * 256 + t] - mean; sq = fmaf(d, d, sq); }
  float var = block_reduce_sum(sq, red, t, 256) * (1.0f / 1024.0f);
  float rstd = rsqrtf(var + LN_EPSF);
  for (int j = 0; j < 4; ++j) {
    int i = j * 256 + t;
    out[p * 1024 + i] = (buf[i] - mean) * rstd * g[i] + b[i];
  }
}

// ---------------------------------------------------------------------------
// Generic WMMA GEMM: D[M,N] = act(A[M,K] @ B[K,N] + bias).  One wave per 16x16
// tile; grid = (M/16, N/16).  All indexing kept 32-bit (no v_mul_u64).
// ---------------------------------------------------------------------------
#define ACT_NONE   0
#define ACT_GELU   1
#define ACT_SPSQRT 2   // sqrt(softplus(x) + 1e-8)

__global__ void k_gemm16(const float* __restrict__ A,
                         const float* __restrict__ B,
                         const float* __restrict__ bias,
                         float* __restrict__ D,
                         int K, int act) {
  const int N  = (int)gridDim.y * 16;
  const int m0 = (int)blockIdx.x * 16, n0 = (int)blockIdx.y * 16;
  const int l  = (int)threadIdx.x;      // 32 lanes
  const int mr = l & 15;
  const int kh = (l >> 4) * 2;

  const float* arow = A + (m0 + mr) * K + kh;   // contiguous pair at arow[k]
  const float* bcol = B + kh * N + (n0 + mr);   // stride-N pair at bcol[k*N]
  v8f acc = {};
  for (int k = 0; k < K; k += 4) {
    v2f a, bb;
    a.x  = arow[k];
    a.y  = arow[k + 1];
    bb.x = bcol[k * N];
    bb.y = bcol[k * N + N];
    acc = wmma_f32_16x16x4(a, bb, acc);
  }
  const int n = n0 + mr;
  const int mbase = m0 + ((l >> 4) ? 8 : 0);
  const float bn = bias[n];
  for (int r = 0; r < 8; ++r) {
    float v = acc[r] + bn;
    if (act == ACT_GELU) {
      v = gelu_exact(v);
    } else if (act == ACT_SPSQRT) {
      float sp = fmaxf(v, 0.0f) + log1pf(__expf(-fabsf(v)));
      v = sqrtf(sp + EPSF);
    }
    D[(mbase + r) * N + n] = v;
  }
}

// ---------------------------------------------------------------------------
// Stage 4: per-pair cost matrix (WMMA) + 60 Sinkhorn iterations, LDS-resident.
// One block per (q,w), 128 threads (4 waves). Dynamic LDS ~71.7KB ->
// 4 blocks/WGP (287KB of the 320KB pool), 16 waves/WGP for the exp-bound loop.
// ---------------------------------------------------------------------------
__global__ void k_sinkhorn(const float* __restrict__ qatoms,  // [256,128,128]
                           const float* __restrict__ catoms,  // [10,128,128]
                           const float* __restrict__ qmass,   // [256,128]
                           const float* __restrict__ cmass,   // [10,128]
                           const float* __restrict__ ssqrt,   // [2560,128] sqrt(scales)
                           const int*   __restrict__ tempp,
                           float* __restrict__ out) {         // [256,10]
  extern __shared__ float smem[];
  float* Cc  = smem;              // 128*129  cost, stride 129 (conflict-free both ways)
  float* sv  = Cc + 128 * 129;    // 128  sqrt(scale)
  float* s2  = sv  + 128;         // 128  scale = sv^2  (folded into A side of GEMM)
  float* rnx = s2  + 128;         // 128  1/||x_m||
  float* x2m = rnx + 128;
  float* rny = x2m + 128;
  float* y2n = rny + 128;
  float* fv  = y2n + 128;
  float* gvv = fv  + 128;
  float* lmu = gvv + 128;
  float* lnu = lmu + 128;
  float* red = lnu + 128;         // 128

  const int p = blockIdx.x, t = threadIdx.x;   // 128 threads
  const int q = p / 10, w = p % 10;

  {
    float s = ssqrt[p * 128 + t];
    sv[t] = s; s2[t] = s * s;
  }

  // normalized masses -> log_mu / log_nu
  float mraw = fmaxf(qmass[q * 128 + t], 0.0f);
  float msum = block_reduce_sum(mraw, red, t, 128);
  lmu[t] = __logf(fmaxf(mraw / fmaxf(msum, EPSF), EPSF));
  float nraw = fmaxf(cmass[w * 128 + t], 0.0f);
  float nsum = block_reduce_sum(nraw, red, t, 128);
  lnu[t] = __logf(fmaxf(nraw / fmaxf(nsum, EPSF), EPSF));
  __syncthreads();

  // row norms of scaled atoms (thread t = row), float4 (b128) loads
  {
    const float4* xr = (const float4*)(qatoms + (q * 128 + t) * 128);
    const float4* yr = (const float4*)(catoms + (w * 128 + t) * 128);
    const float4* s4 = (const float4*)sv;
    float ax = 0.0f, ay = 0.0f;
    for (int d = 0; d < 32; ++d) {
      float4 s = s4[d];
      float4 xv = xr[d], yv = yr[d];
      float x0 = xv.x * s.x, x1 = xv.y * s.y, x2 = xv.z * s.z, x3 = xv.w * s.w;
      float y0 = yv.x * s.x, y1 = yv.y * s.y, y2 = yv.z * s.z, y3 = yv.w * s.w;
      ax = fmaf(x0, x0, fmaf(x1, x1, fmaf(x2, x2, fmaf(x3, x3, ax))));
      ay = fmaf(y0, y0, fmaf(y1, y1, fmaf(y2, y2, fmaf(y3, y3, ay))));
    }
    float nx = sqrtf(ax), ny = sqrtf(ay);
    float rx = 1.0f / fmaxf(nx, EPSF), ry = 1.0f / fmaxf(ny, EPSF);
    rnx[t] = rx; x2m[t] = (nx * rx) * (nx * rx);
    rny[t] = ry; y2n[t] = (ny * ry) * (ny * ry);
  }
  __syncthreads();

  // Cost matrix via WMMA: dot(x*s, y*s) == dot(x*s^2, y); norms applied in
  // the epilogue -> B fragment is a raw contiguous b64 load (no VALU).
  {
    const int wave = t >> 5, lane = t & 31;
    const int mr = lane & 15;
    const int kh = (lane >> 4) * 2;
    for (int i = 0; i < 16; ++i) {
      const int tile = i * 4 + wave;       // 64 tiles, 8x8 of 16x16
      const int m0 = (tile >> 3) << 4;
      const int n0 = (tile & 7) << 4;
      const float* ax = qatoms + (q * 128 + m0 + mr) * 128 + kh;
      const float* by = catoms + (w * 128 + n0 + mr) * 128 + kh;
      v8f acc = {};
      for (int k = 0; k < 128; k += 4) {
        v2f a, bb;
        a.x  = ax[k]     * s2[k + kh];
        a.y  = ax[k + 1] * s2[k + kh + 1];
        bb.x = by[k];
        bb.y = by[k + 1];
        acc = wmma_f32_16x16x4(a, bb, acc);
      }
      const int n = n0 + mr;
      const int mbase = m0 + ((lane >> 4) ? 8 : 0);
      const float ryy = rny[n], yy = y2n[n];
      for (int r = 0; r < 8; ++r) {
        const int m = mbase + r;
        float c = x2m[m] + yy - 2.0f * acc[r] * rnx[m] * ryy;
        Cc[m * 129 + n] = fmaxf(c, 0.0f);
      }
    }
  }
  fv[t] = 0.0f; gvv[t] = 0.0f;
  __syncthreads();

  // Sinkhorn: stabilized LSE, 4-way accumulators to break FP dep chains,
  // exp2/log2 with folded constants (v_exp_f32 is natively 2^x).
  for (int it = 0; it < SINK_ITERS; ++it) {
    {   // f-update: thread t = row m
      const float* crow = Cc + t * 129;
      float m0 = -3.4e38f, m1 = -3.4e38f, m2 = -3.4e38f, m3 = -3.4e38f;
      for (int n = 0; n < 128; n += 4) {
        m0 = fmaxf(m0, gvv[n]     - crow[n]);
        m1 = fmaxf(m1, gvv[n + 1] - crow[n + 1]);
        m2 = fmaxf(m2, gvv[n + 2] - crow[n + 2]);
        m3 = fmaxf(m3, gvv[n + 3] - crow[n + 3]);
      }
      float mx = fmaxf(fmaxf(m0, m1), fmaxf(m2, m3));
      float e0 = 0.0f, e1 = 0.0f, e2 = 0.0f, e3 = 0.0f;
      for (int n = 0; n < 128; n += 4) {
        e0 += exp2f((gvv[n]     - crow[n]     - mx) * K2E);
        e1 += exp2f((gvv[n + 1] - crow[n + 1] - mx) * K2E);
        e2 += exp2f((gvv[n + 2] - crow[n + 2] - mx) * K2E);
        e3 += exp2f((gvv[n + 3] - crow[n + 3] - mx) * K2E);
      }
      float se = (e0 + e1) + (e2 + e3);
      fv[t] = SINK_EPS * lmu[t] - mx - CLG * __log2f(se);
    }
    __syncthreads();
    {   // g-update: thread t = column n
      const float* ccol = Cc + t;
      float m0 = -3.4e38f, m1 = -3.4e38f, m2 = -3.4e38f, m3 = -3.4e38f;
      for (int m = 0; m < 128; m += 4) {
        m0 = fmaxf(m0, fv[m]     - ccol[m * 129]);
        m1 = fmaxf(m1, fv[m + 1] - ccol[(m + 1) * 129]);
        m2 = fmaxf(m2, fv[m + 2] - ccol[(m + 2) * 129]);
        m3 = fmaxf(m3, fv[m + 3] - ccol[(m + 3) * 129]);
      }
      float mx = fmaxf(fmaxf(m0, m1), fmaxf(m2, m3));
      float e0 = 0.0f, e1 = 0.0f, e2 = 0.0f, e3 = 0.0f;
      for (int m = 0; m < 128; m += 4) {
        e0 += exp2f((fv[m]     - ccol[m * 129]       - mx) * K2E);
        e1 += exp2f((fv[m + 1] - ccol[(m + 1) * 129] - mx) * K2E);
        e2 += exp2f((fv[m + 2] - ccol[(m + 2) * 129] - mx) * K2E);
        e3 += exp2f((fv[m + 3] - ccol[(m + 3) * 129] - mx) * K2E);
      }
      float se = (e0 + e1) + (e2 + e3);
      gvv[t] = SINK_EPS * lnu[t] - mx - CLG * __log2f(se);
    }
    __syncthreads();
  }

  // dist = sum plan * C,  plan = exp((f+g-C)/eps)
  {
    const float* crow = Cc + t * 129;
    const float ft = fv[t];
    float e0 = 0.0f, e1 = 0.0f, e2 = 0.0f, e3 = 0.0f;
    for (int n = 0; n < 128; n += 4) {
      float c0 = crow[n], c1 = crow[n + 1], c2 = crow[n + 2], c3 = crow[n + 3];
      e0 = fmaf(exp2f((ft + gvv[n]     - c0) * K2E), c0, e0);
      e1 = fmaf(exp2f((ft + gvv[n + 1] - c1) * K2E), c1, e1);
      e2 = fmaf(exp2f((ft + gvv[n + 2] - c2) * K2E), c2, e2);
      e3 = fmaf(exp2f((ft + gvv[n + 3] - c3) * K2E), c3, e3);
    }
    float dist = block_reduce_sum((e0 + e1) + (e2 + e3), red, t, 128);
    if (t == 0) out[q * 10 + w] = -(float)tempp[0] * dist;
  }
}

// ---------------------------------------------------------------------------
extern "C" void kernel_launch(void* const* d_in, const int* in_sizes, int n_in,
                              void* d_out, int out_size, void* d_ws, size_t ws_size,
                              hipStream_t stream) {
  const float* qatoms = (const float*)d_in[0];
  const float* qmass  = (const float*)d_in[1];
  const float* catoms = (const float*)d_in[2];
  const float* cmass  = (const float*)d_in[3];
  const float* csum   = (const float*)d_in[4];
  const float* ectx   = (const float*)d_in[5];
  const float* qp_g  = (const float*)d_in[6],  *qp_b  = (const float*)d_in[7];
  const float* qp_w1 = (const float*)d_in[8],  *qp_b1 = (const float*)d_in[9];
  const float* qp_w2 = (const float*)d_in[10], *qp_b2 = (const float*)d_in[11];
  const float* cp_g  = (const float*)d_in[12], *cp_b  = (const float*)d_in[13];
  const float* cp_w1 = (const float*)d_in[14], *cp_b1 = (const float*)d_in[15];
  const float* cp_w2 = (const float*)d_in[16], *cp_b2 = (const float*)d_in[17];
  const float* mh_g  = (const float*)d_in[18], *mh_b  = (const float*)d_in[19];
  const float* mh_w1 = (const float*)d_in[20], *mh_b1 = (const float*)d_in[21];
  const float* mh_w2 = (const float*)d_in[22], *mh_b2 = (const float*)d_in[23];
  const int*   temp  = (const int*)d_in[24];
  float* out = (float*)d_out;
  float* ws  = (float*)d_ws;

  // workspace layout (floats), total ~15.1 MB
  float* pooled = ws;                     // 256*128
  float* hq     = pooled + 256 * 128;     // 256*256
  float* qh     = hq     + 256 * 256;     // 256*256
  float* ctxln  = qh     + 256 * 256;     // 16*512
  float* hc     = ctxln  + 16 * 512;      // 16*256
  float* ch     = hc     + 16 * 256;      // 16*256
  float* fused  = ch     + 16 * 256;      // 2560*1024
  float* hm     = fused  + 2560 * 1024;   // 2560*256
  float* ssq    = hm     + 2560 * 256;    // 2560*128

  // query projection
  k_pool<<<256, 128, 0, stream>>>(qatoms, qmass, qp_g, qp_b, pooled);
  k_gemm16<<<dim3(16, 16), 32, 0, stream>>>(pooled, qp_w1, qp_b1, hq, 128, ACT_GELU);
  k_gemm16<<<dim3(16, 16), 32, 0, stream>>>(hq, qp_w2, qp_b2, qh, 256, ACT_NONE);

  // context projection (rows padded to 16)
  k_ctx<<<16, 512, 0, stream>>>(csum, ectx, cp_g, cp_b, ctxln);
  k_gemm16<<<dim3(1, 16), 32, 0, stream>>>(ctxln, cp_w1, cp_b1, hc, 512, ACT_GELU);
  k_gemm16<<<dim3(1, 16), 32, 0, stream>>>(hc, cp_w2, cp_b2, ch, 256, ACT_NONE);

  // metric head -> sqrt(softplus(.) + eps)
  k_fused<<<2560, 256, 0, stream>>>(qh, ch, mh_g, mh_b, fused);
  k_gemm16<<<dim3(160, 16), 32, 0, stream>>>(fused, mh_w1, mh_b1, hm, 1024, ACT_GELU);
  k_gemm16<<<dim3(160, 8), 32, 0, stream>>>(hm, mh_w2, mh_b2, ssq, 256, ACT_SPSQRT);

  // per-pair optimal transport
  const size_t lds = (size_t)(128 * 129 + 11 * 128) * sizeof(float);  // 71.7KB
  k_sinkhorn<<<2560, 128, lds, stream>>>(qatoms, catoms, qmass, cmass, ssq, temp, out);
}